// model_60627758351009
// MI455X (gfx1250) — compile-verified
//
#include <hip/hip_runtime.h>
#include <hip/hip_bf16.h>

#define EPS       1e-10f
#define DIM       256
#define NUM_SETS  8000
#define MAX_SET   50
#define BATCHSZ   8192
#define KPAD      52        // MAX_SET padded to multiple of 4 for WMMA K-tiling
#define NKT       (KPAD/4)  // 13 K-tiles of 4
#define XSTRIDE   264       // LDS row stride (floats), multiple of 4 for float4 stores

typedef float v2f __attribute__((ext_vector_type(2)));
typedef float v8f __attribute__((ext_vector_type(8)));

__device__ __forceinline__ float wave_sum(float v) {
    #pragma unroll
    for (int off = 16; off; off >>= 1) v += __shfl_xor(v, off);
    return v;
}
__device__ __forceinline__ float wave_max(float v) {
    #pragma unroll
    for (int off = 16; off; off >>= 1) v = fmaxf(v, __shfl_xor(v, off));
    return v;
}

// log(softplus(x) + EPS), softplus computed stably (BETA = 1)
__device__ __forceinline__ float lvterm(float x) {
    float sp = fmaxf(x, 0.0f) + log1pf(expf(-fabsf(x)));
    return logf(sp + EPS);
}

// ---------------------------------------------------------------------------
// Kernel 1: set-attention pooling.  grid = (NUM_SETS, 2), block = 256.
//   blockIdx.y == 0 : center table (no size regularization)
//   blockIdx.y == 1 : radius table (emb *= size^(1/DIM))
// ---------------------------------------------------------------------------
__global__ __launch_bounds__(256)
void attention_scp_kernel(const int*   __restrict__ S,
                          const float* __restrict__ M,
                          const float* __restrict__ cemb,
                          const float* __restrict__ remb,
                          const float* __restrict__ catt,
                          const float* __restrict__ ratt,
                          float*       __restrict__ out_c,
                          float*       __restrict__ out_r)
{
    __shared__ float Xg[KPAD * XSTRIDE];   // gathered 50x256 tile (+pad rows/cols)
    __shared__ float attl[DIM];
    __shared__ float avec[DIM];
    __shared__ float logits[64];
    __shared__ float wvec[64];
    __shared__ float msk[64];
    __shared__ float ssize;

    const int set   = blockIdx.x;
    const int table = blockIdx.y;
    const int tid   = threadIdx.x;
    const int wave  = tid >> 5;
    const int lane  = tid & 31;
    const int hlf   = lane >> 4;   // which half of the wave (K pair select)
    const int nl    = lane & 15;   // N column within tile / M row for A-frag

    const float* X   = (table == 0) ? cemb : remb;
    const float* A   = (table == 0) ? catt : ratt;
    float*       dst = (table == 0) ? out_c : out_r;

    // --- stage in attention vector + mask ---
    attl[tid] = A[tid];
    if (tid < 64) msk[tid] = (tid < MAX_SET) ? M[set * MAX_SET + tid] : 0.0f;

    // --- gather Xg rows into LDS (float4, coalesced: 4 rows x 64 lanes) ---
    {
        const int rgrp = tid >> 6;            // 0..3
        const int c4   = (tid & 63) << 2;     // float4 column start
        for (int base = 0; base < MAX_SET; base += 4) {
            int r = base + rgrp;
            if (r < MAX_SET) {
                int idx = S[set * MAX_SET + r];
                float4 v = *(const float4*)(X + (size_t)idx * DIM + c4);
                *(float4*)(&Xg[r * XSTRIDE + c4]) = v;
            }
        }
        // zero WMMA K-padding rows (50,51) so padded K contributes exactly 0
        #pragma unroll
        for (int r = MAX_SET; r < KPAD; ++r) Xg[r * XSTRIDE + tid] = 0.0f;
    }
    __syncthreads();

    // set size = sum(mask)
    if (tid < 32) {
        float s = wave_sum(msk[lane] + msk[lane + 32]);
        if (lane == 0) ssize = s;
    }

    // =============== pass 1: logits = Xg @ att ===============
    for (int r = wave; r < MAX_SET; r += 8) {
        float p = 0.0f;
        #pragma unroll
        for (int j = 0; j < 8; ++j) {
            int d = lane + 32 * j;
            p += Xg[r * XSTRIDE + d] * attl[d];
        }
        p = wave_sum(p);
        if (lane == 0) logits[r] = p;
    }
    __syncthreads();

    // =============== masked softmax -> wvec (wave 0) ===============
    if (tid < 32) {
        float l0 = (lane < MAX_SET      && msk[lane]      > 0.0f) ? logits[lane]      : -1e30f;
        float l1 = (lane + 32 < MAX_SET && msk[lane + 32] > 0.0f) ? logits[lane + 32] : -1e30f;
        float mx = wave_max(fmaxf(l0, l1));
        float e0 = (l0 > -1e29f) ? expf(l0 - mx) : 0.0f;
        float e1 = (l1 > -1e29f) ? expf(l1 - mx) : 0.0f;
        float s  = wave_sum(e0 + e1);
        wvec[lane]      = e0 / s;
        wvec[lane + 32] = e1 / s;
    }
    __syncthreads();

    // =============== a = w1^T Xg via V_WMMA_F32_16X16X4_F32 ===============
    // A 16x4 layout: lanes 0-15 hold M=0..15 with K=0,1; lanes 16-31 same M, K=2,3.
    // Only M-row 0 is populated -> lanes with nl==0 carry the weights.
    // A-fragment is nt-invariant: preload all NKT fragments once per wave.
    // Loads are UNCONDITIONAL (b64 pair) + value-select, so EXEC stays full
    // and no saveexec/masked-load sequences appear around the WMMAs.
    {
        v2f afrag[NKT];
        #pragma unroll
        for (int kt = 0; kt < NKT; ++kt) {
            float2 wp = *(const float2*)&wvec[4 * kt + 2 * hlf];   // ds_load_b64
            afrag[kt].x = (nl == 0) ? wp.x : 0.0f;                 // v_cndmask
            afrag[kt].y = (nl == 0) ? wp.y : 0.0f;
        }
        for (int nt = wave; nt < 16; nt += 8) {
            const int n = nt * 16 + nl;
            v8f c = {};
            #pragma unroll
            for (int kt = 0; kt < NKT; ++kt) {
                const int k0 = 4 * kt + 2 * hlf;
                v2f bv = {Xg[k0 * XSTRIDE + n], Xg[(k0 + 1) * XSTRIDE + n]};
                c = __builtin_amdgcn_wmma_f32_16x16x4_f32(
                        false, afrag[kt], false, bv, (short)0, c, false, false);
            }
            if (hlf == 0) avec[n] = c[0];    // D row 0 = lanes 0-15 of VGPR0
        }
    }
    __syncthreads();

    // =============== pass 2: logits2 = Xg @ a ===============
    for (int r = wave; r < MAX_SET; r += 8) {
        float p = 0.0f;
        #pragma unroll
        for (int j = 0; j < 8; ++j) {
            int d = lane + 32 * j;
            p += Xg[r * XSTRIDE + d] * avec[d];
        }
        p = wave_sum(p);
        if (lane == 0) logits[r] = p;
    }
    __syncthreads();

    // =============== masked softmax -> wvec (w2) ===============
    if (tid < 32) {
        float l0 = (lane < MAX_SET      && msk[lane]      > 0.0f) ? logits[lane]      : -1e30f;
        float l1 = (lane + 32 < MAX_SET && msk[lane + 32] > 0.0f) ? logits[lane + 32] : -1e30f;
        float mx = wave_max(fmaxf(l0, l1));
        float e0 = (l0 > -1e29f) ? expf(l0 - mx) : 0.0f;
        float e1 = (l1 > -1e29f) ? expf(l1 - mx) : 0.0f;
        float s  = wave_sum(e0 + e1);
        wvec[lane]      = e0 / s;
        wvec[lane + 32] = e1 / s;
    }
    __syncthreads();

    // =============== emb = w2^T Xg via WMMA, write to global ===============
    {
        const float scale = (table == 1) ? expf(logf(ssize) * (1.0f / (float)DIM)) : 1.0f;
        v2f afrag[NKT];
        #pragma unroll
        for (int kt = 0; kt < NKT; ++kt) {
            float2 wp = *(const float2*)&wvec[4 * kt + 2 * hlf];
            afrag[kt].x = (nl == 0) ? wp.x : 0.0f;
            afrag[kt].y = (nl == 0) ? wp.y : 0.0f;
        }
        for (int nt = wave; nt < 16; nt += 8) {
            const int n = nt * 16 + nl;
            v8f c = {};
            #pragma unroll
            for (int kt = 0; kt < NKT; ++kt) {
                const int k0 = 4 * kt + 2 * hlf;
                v2f bv = {Xg[k0 * XSTRIDE + n], Xg[(k0 + 1) * XSTRIDE + n]};
                c = __builtin_amdgcn_wmma_f32_16x16x4_f32(
                        false, afrag[kt], false, bv, (short)0, c, false, false);
            }
            if (hlf == 0) dst[(size_t)set * DIM + n] = c[0] * scale;
        }
    }
}

// ---------------------------------------------------------------------------
// Kernel 2: box-volume terms + per-row softmax + squared error.
// One wave per batch row; block = 256 (8 rows), grid = BATCH/8.
// ---------------------------------------------------------------------------
__global__ __launch_bounds__(256)
void boxvol_kernel(const int*   __restrict__ inst,
                   const float* __restrict__ ov,
                   const float* __restrict__ emb_c,
                   const float* __restrict__ emb_r,
                   float*       __restrict__ sq)
{
    const int wave = threadIdx.x >> 5;
    const int lane = threadIdx.x & 31;
    const int b    = blockIdx.x * 8 + wave;
    if (b >= BATCHSZ) return;

    const size_t o0 = (size_t)inst[b * 3 + 0] * DIM;
    const size_t o1 = (size_t)inst[b * 3 + 1] * DIM;
    const size_t o2 = (size_t)inst[b * 3 + 2] * DIM;

    float acc[7];
    #pragma unroll
    for (int t = 0; t < 7; ++t) acc[t] = 0.0f;

    #pragma unroll
    for (int j = 0; j < 8; ++j) {
        int d = lane + 32 * j;
        float ci = emb_c[o0 + d], cj = emb_c[o1 + d], ck = emb_c[o2 + d];
        float ri = emb_r[o0 + d], rj = emb_r[o1 + d], rk = emb_r[o2 + d];
        float mi = ci - ri, Mi = ci + ri;
        float mj = cj - rj, Mj = cj + rj;
        float mk = ck - rk, Mk = ck + rk;
        acc[0] += lvterm(Mi - mi);
        acc[1] += lvterm(Mj - mj);
        acc[2] += lvterm(Mk - mk);
        acc[3] += lvterm(fminf(Mi, Mj) - fmaxf(mi, mj));
        acc[4] += lvterm(fminf(Mj, Mk) - fmaxf(mj, mk));
        acc[5] += lvterm(fminf(Mk, Mi) - fmaxf(mk, mi));
        acc[6] += lvterm(fminf(Mi, fminf(Mj, Mk)) - fmaxf(mi, fmaxf(mj, mk)));
    }
    #pragma unroll
    for (int t = 0; t < 7; ++t) acc[t] = wave_sum(acc[t]);

    if (lane == 0) {
        // pred = softmax(terms) : the reference's global Z cancels exactly,
        // so the per-row max gives the identical result with better stability.
        float mx = acc[0];
        #pragma unroll
        for (int t = 1; t < 7; ++t) mx = fmaxf(mx, acc[t]);
        float e[7], s = 0.0f;
        #pragma unroll
        for (int t = 0; t < 7; ++t) { e[t] = expf(acc[t] - mx); s += e[t]; }
        float inv = 1.0f / s, err = 0.0f;
        #pragma unroll
        for (int t = 0; t < 7; ++t) {
            float dd = e[t] * inv - ov[b * 7 + t];
            err += dd * dd;
        }
        sq[b] = err;
    }
}

// ---------------------------------------------------------------------------
// Kernel 3: deterministic tree reduction of 8192 partials -> scalar.
// ---------------------------------------------------------------------------
__global__ __launch_bounds__(256)
void reduce_kernel(const float* __restrict__ sq, float* __restrict__ out)
{
    __shared__ float red[256];
    const int tid = threadIdx.x;
    float s = 0.0f;
    for (int i = tid; i < BATCHSZ; i += 256) s += sq[i];
    red[tid] = s;
    __syncthreads();
    #pragma unroll
    for (int off = 128; off; off >>= 1) {
        if (tid < off) red[tid] += red[tid + off];
        __syncthreads();
    }
    if (tid == 0) out[0] = red[0];
}

// ---------------------------------------------------------------------------
extern "C" void kernel_launch(void* const* d_in, const int* in_sizes, int n_in,
                              void* d_out, int out_size, void* d_ws, size_t ws_size,
                              hipStream_t stream)
{
    const int*   S    = (const int*)  d_in[0];
    const float* M    = (const float*)d_in[1];
    const int*   inst = (const int*)  d_in[2];
    const float* ov   = (const float*)d_in[3];
    const float* cemb = (const float*)d_in[4];
    const float* remb = (const float*)d_in[5];
    const float* catt = (const float*)d_in[6];
    const float* ratt = (const float*)d_in[7];
    float* out = (float*)d_out;

    float* ws    = (float*)d_ws;
    float* emb_c = ws;
    float* emb_r = ws + (size_t)NUM_SETS * DIM;
    float* sq    = ws + (size_t)2 * NUM_SETS * DIM;

    attention_scp_kernel<<<dim3(NUM_SETS, 2), 256, 0, stream>>>(
        S, M, cemb, remb, catt, ratt, emb_c, emb_r);
    boxvol_kernel<<<BATCHSZ / 8, 256, 0, stream>>>(inst, ov, emb_c, emb_r, sq);
    reduce_kernel<<<1, 256, 0, stream>>>(sq, out);
}